// NN_split_70592082477596
// MI455X (gfx1250) — compile-verified
//
#include <hip/hip_runtime.h>
#include <stdint.h>

#define B_DIM 256
#define N_DIM 20000
#define G_DIM 4096
#define L_DIM 128
#define KCHUNKS 8
#define KSPAN (G_DIM / KCHUNKS)   // 512

typedef float v2f __attribute__((ext_vector_type(2)));
typedef float v8f __attribute__((ext_vector_type(8)));

// ---------------------------------------------------------------------------
// Kernel A: transpose x (B=256, N=20000) -> xT (N, B) so a gathered "column"
// of x becomes a contiguous 1KB row (coalesced b128 loads in kernel B).
// Grid: (20000/32, 256/32) = (625, 8); block (32, 8), 4 rows per thread.
// ---------------------------------------------------------------------------
__global__ __launch_bounds__(256) void k_transpose(const float* __restrict__ x,
                                                   float* __restrict__ xT) {
  __shared__ float tile[32][33];  // +1 pad: conflict-free transpose read
  const int tx = threadIdx.x, ty = threadIdx.y;
  const int n0 = blockIdx.x * 32, b0 = blockIdx.y * 32;
#pragma unroll
  for (int j = 0; j < 4; ++j) {
    const int b = ty + j * 8;
    tile[b][tx] = x[(size_t)(b0 + b) * N_DIM + (n0 + tx)];
  }
  __syncthreads();
#pragma unroll
  for (int j = 0; j < 4; ++j) {
    const int r = ty + j * 8;
    xT[(size_t)(n0 + r) * B_DIM + (b0 + tx)] = tile[tx][r];
  }
}

// ---------------------------------------------------------------------------
// Kernel B: hidden[g, b] = relu( sum_l wm[g,l] * xT[idx[g,l], b] + bias[g] )
// Block = 256 threads = 8 waves = 4 groups x 2 batch-halves.
// Each lane holds a float4 accumulator (4 batch entries); each wave reads a
// contiguous 512B slice of every gathered 1KB row. Masked weights + indices
// staged in LDS (uniform ds_load broadcast in the loop). global_prefetch_b8
// lookahead (16 rows) hides L2 gather latency.
// hidden stored as (G, B): coalesced stores here, contiguous B-matrix rows
// for the WMMA kernel.
// ---------------------------------------------------------------------------
__global__ __launch_bounds__(256) void k_einsum(const float* __restrict__ xT,
                                                const int* __restrict__ idx,
                                                const unsigned char* __restrict__ mask,
                                                const float* __restrict__ w,
                                                const float* __restrict__ bias,
                                                float* __restrict__ hid) {
  __shared__ float sW[4 * L_DIM];
  __shared__ int   sI[4 * L_DIM];
  const int tid = threadIdx.x;
  const int gBase = blockIdx.x * 4;

  for (int i = tid; i < 4 * L_DIM; i += 256) {
    const int gl = gBase * L_DIM + i;
    sW[i] = mask[gl] ? w[gl] : 0.0f;   // fold mask into the weight once
    sI[i] = idx[gl];
  }
  __syncthreads();

  const int wid = tid >> 5, lane = tid & 31;
  const int gg = wid >> 1;            // group within block (0..3)
  const int bhalf = wid & 1;          // which 128-batch half
  const int g = gBase + gg;
  const int boff = bhalf * 128 + lane * 4;   // 4 consecutive batch entries

  const float* __restrict__ sWg = &sW[gg * L_DIM];
  const int*   __restrict__ sIg = &sI[gg * L_DIM];

  float ax = 0.0f, ay = 0.0f, az = 0.0f, aw = 0.0f;

#pragma unroll 4
  for (int l = 0; l < L_DIM; ++l) {
    if (l + 16 < L_DIM) {  // prefetch the gathered row 16 iterations ahead
      __builtin_prefetch(xT + (size_t)sIg[l + 16] * B_DIM + boff, 0, 1);
    }
    const float  wv = sWg[l];
    const float4 xv = *reinterpret_cast<const float4*>(
        xT + (size_t)sIg[l] * B_DIM + boff);
    ax = fmaf(wv, xv.x, ax);
    ay = fmaf(wv, xv.y, ay);
    az = fmaf(wv, xv.z, az);
    aw = fmaf(wv, xv.w, aw);
  }

  const float bb = bias[g];
  float4 h;
  h.x = fmaxf(ax + bb, 0.0f);
  h.y = fmaxf(ay + bb, 0.0f);
  h.z = fmaxf(az + bb, 0.0f);
  h.w = fmaxf(aw + bb, 0.0f);
  *reinterpret_cast<float4*>(hid + (size_t)g * B_DIM + boff) = h;
}

// ---------------------------------------------------------------------------
// Kernel C: partial logits via V_WMMA_F32_16X16X4_F32 (exact fp32).
// D[k, b] = sum_{g in chunk} fc2_w[k, g] * hid[g, b].
// Grid: (16 batch tiles) x (8 K-chunks of 512), one wave each -> 128 waves.
//   A (16x4): lane m = lane&15 is row M; half = lane>>4 selects K = 2h+{0,1}.
//             Rows 2..15 must be zero (only 2 classes): load from a CLAMPED
//             valid row unconditionally, then v_cndmask to zero -> no exec-
//             mask branching inside the WMMA loop.
//   B (4x16): lane n = lane&15 is column N (batch); same K mapping.
//             hid is (G, B) so B rows are contiguous 64B half-wave loads.
//   D: c[0] lanes 0-15 = partial class-0 logits, c[1] lanes 0-15 = class 1.
// Partials are written per (chunk, b) and summed later in a FIXED order so
// results are deterministic across replays (no float atomics).
// ---------------------------------------------------------------------------
__global__ __launch_bounds__(32) void k_fc2_partial(
    const float* __restrict__ hid, const float* __restrict__ fc2w,
    float* __restrict__ part) {
  const int lane = threadIdx.x;
  const int m = lane & 15;
  const int half = lane >> 4;
  const int bbase = blockIdx.x * 16;
  const int kbase = blockIdx.y * KSPAN;

  const bool valid = (m < 2);
  const float* __restrict__ arow = fc2w + (size_t)(valid ? m : 0) * G_DIM;

  v8f c = {0.f, 0.f, 0.f, 0.f, 0.f, 0.f, 0.f, 0.f};

#pragma unroll 4
  for (int k0 = kbase; k0 < kbase + KSPAN; k0 += 4) {
    const int kA = k0 + 2 * half;
    const float a0 = arow[kA];          // unconditional load (clamped row)
    const float a1 = arow[kA + 1];
    v2f A, Bv;
    A.x = valid ? a0 : 0.0f;            // branchless v_cndmask
    A.y = valid ? a1 : 0.0f;
    Bv.x = hid[(size_t)kA * B_DIM + bbase + m];
    Bv.y = hid[(size_t)(kA + 1) * B_DIM + bbase + m];
    // (neg_a, A, neg_b, B, c_mod, C, reuse_a, reuse_b)
    c = __builtin_amdgcn_wmma_f32_16x16x4_f32(false, A, false, Bv,
                                              (short)0, c, false, false);
  }

  if (lane < 16) {  // layout: part[kchunk][b][class]
    float* p = part + (size_t)blockIdx.y * (B_DIM * 2) + (size_t)(bbase + m) * 2;
    p[0] = c[0];
    p[1] = c[1];
  }
}

// ---------------------------------------------------------------------------
// Kernel D: sum the 8 K-chunk partials in fixed order, add fc2_b, log-softmax
// over the 2 classes. One thread per batch element.
// ---------------------------------------------------------------------------
__global__ __launch_bounds__(256) void k_logsoftmax(
    const float* __restrict__ part, const float* __restrict__ fc2b,
    float* __restrict__ out) {
  const int b = threadIdx.x;
  float l0 = fc2b[0];
  float l1 = fc2b[1];
#pragma unroll
  for (int kc = 0; kc < KCHUNKS; ++kc) {
    l0 += part[(size_t)kc * (B_DIM * 2) + b * 2 + 0];
    l1 += part[(size_t)kc * (B_DIM * 2) + b * 2 + 1];
  }
  const float mx = fmaxf(l0, l1);
  const float lse = mx + logf(expf(l0 - mx) + expf(l1 - mx));
  out[(size_t)b * 2 + 0] = l0 - lse;
  out[(size_t)b * 2 + 1] = l1 - lse;
}

// ---------------------------------------------------------------------------
extern "C" void kernel_launch(void* const* d_in, const int* in_sizes, int n_in,
                              void* d_out, int out_size, void* d_ws, size_t ws_size,
                              hipStream_t stream) {
  const float*         x    = (const float*)d_in[0];
  const int*           idx  = (const int*)d_in[1];
  const unsigned char* mask = (const unsigned char*)d_in[2];  // jnp bool_ = 1 byte
  const float*         w    = (const float*)d_in[3];
  const float*         bias = (const float*)d_in[4];
  const float*         fc2w = (const float*)d_in[5];
  const float*         fc2b = (const float*)d_in[6];
  float* out = (float*)d_out;

  char* ws = (char*)d_ws;
  float* xT   = (float*)ws;                                        // 20.48 MB
  float* hid  = (float*)(ws + (size_t)N_DIM * B_DIM * 4);          // + 4 MB
  float* part = (float*)(ws + (size_t)N_DIM * B_DIM * 4
                            + (size_t)G_DIM * B_DIM * 4);          // + 16 KB

  k_transpose<<<dim3(N_DIM / 32, B_DIM / 32), dim3(32, 8), 0, stream>>>(x, xT);
  k_einsum<<<G_DIM / 4, 256, 0, stream>>>(xT, idx, mask, w, bias, hid);
  k_fc2_partial<<<dim3(B_DIM / 16, KCHUNKS), 32, 0, stream>>>(hid, fc2w, part);
  k_logsoftmax<<<1, B_DIM, 0, stream>>>(part, fc2b, out);
}